// RNNTLoss_17849884082668
// MI455X (gfx1250) — compile-verified
//
#include <hip/hip_runtime.h>
#include <hip/hip_bf16.h>
#include <math.h>
#include <stdint.h>

// Problem constants (match reference).
#define B_ 8
#define T_ 256
#define U_ 64
#define V_ 512

#define NEG_INF (-__builtin_inff())

// Native clang vector type: keeps everything in VGPRs (no HIP float4 class
// aggregates, which round 2 showed clang lowering to a scratch alloca).
typedef float rnnt_v4f __attribute__((ext_vector_type(4)));

// ---------------------------------------------------------------------------
// CDNA5 async global->LDS path (gfx1250). Builtin signature (probe-confirmed):
//   void __builtin_amdgcn_global_load_async_to_lds_b128(
//       int __vector(4) addrspace(1)* gsrc, int __vector(4) addrspace(3)* ldst,
//       imm int offset, imm int cpol)
// Flat->AS conversions via integer round-trip: global flat == 64-bit VA,
// LDS flat carries the LDS byte offset in its low 32 bits (ISA aperture rule).
// ---------------------------------------------------------------------------
#if defined(__HIP_DEVICE_COMPILE__) && \
    __has_builtin(__builtin_amdgcn_global_load_async_to_lds_b128) && \
    __has_builtin(__builtin_amdgcn_s_wait_asynccnt)
#define RNNT_ASYNC_LDS 1
typedef int rnnt_v4i __attribute__((vector_size(16)));
typedef __attribute__((address_space(1))) rnnt_v4i* rnnt_gas_v4i;
typedef __attribute__((address_space(3))) rnnt_v4i* rnnt_lds_v4i;
#define RNNT_AS1_V4I(p) ((rnnt_gas_v4i)(uintptr_t)(p))
#define RNNT_AS3_V4I(p) ((rnnt_lds_v4i)(unsigned)(uintptr_t)(p))
#else
#define RNNT_ASYNC_LDS 0
#endif

// ---------------------------------------------------------------------------
// Phase 2 (defined FIRST so its asm heads the module and the async-to-LDS
// lowering is visible in the disasm snippet): anti-diagonal wavefront DP,
// one block per batch element.
// alpha[t,u] = logaddexp(alpha[t-1,u] + lpb[t-1,u], alpha[t,u-1] + lpl[t,u-1])
// The whole per-batch lpb/lpl slab (132 KB) fits in CDNA5's 320 KB WGP LDS,
// staged via gfx1250 async global->LDS DMA, so every DP step (320 serial
// diagonals) reads only LDS.
// ---------------------------------------------------------------------------
__global__ __launch_bounds__(128)
void rnnt_dp_kernel(const float* __restrict__ lpb,
                    const float* __restrict__ lpl,
                    const int*   __restrict__ loglen,
                    const int*   __restrict__ tgtlen,
                    float* __restrict__ ll) {
    __shared__ float lpb_s[T_ * (U_ + 1)];   // 66,560 B
    __shared__ float lpl_s[T_ * U_];         // 65,536 B
    __shared__ float alpha[2][U_ + 1];

    const int b   = blockIdx.x;
    const int tid = threadIdx.x;
    const float* gb = lpb + (size_t)b * T_ * (U_ + 1);
    const float* gl = lpl + (size_t)b * T_ * U_;

#if RNNT_ASYNC_LDS
    for (int i = tid; i < T_ * (U_ + 1) / 4; i += 128)
        __builtin_amdgcn_global_load_async_to_lds_b128(
            RNNT_AS1_V4I(gb + i * 4), RNNT_AS3_V4I(lpb_s + i * 4), 0, 0);
    for (int i = tid; i < T_ * U_ / 4; i += 128)
        __builtin_amdgcn_global_load_async_to_lds_b128(
            RNNT_AS1_V4I(gl + i * 4), RNNT_AS3_V4I(lpl_s + i * 4), 0, 0);
    __builtin_amdgcn_s_wait_asynccnt(0);
#else
    {
        const rnnt_v4f* gb4 = reinterpret_cast<const rnnt_v4f*>(gb);
        const rnnt_v4f* gl4 = reinterpret_cast<const rnnt_v4f*>(gl);
        rnnt_v4f* lb4 = reinterpret_cast<rnnt_v4f*>(lpb_s);
        rnnt_v4f* ll4 = reinterpret_cast<rnnt_v4f*>(lpl_s);
        for (int i = tid; i < T_ * (U_ + 1) / 4; i += 128) lb4[i] = gb4[i];
        for (int i = tid; i < T_ * U_ / 4; i += 128) ll4[i] = gl4[i];
    }
#endif
    if (tid <= U_) {
        alpha[0][tid] = (tid == 0) ? 0.0f : NEG_INF;
        alpha[1][tid] = NEG_INF;
    }
    __syncthreads();

    const int te   = loglen[b] - 1;   // final time index
    const int ue   = tgtlen[b];       // final label index
    const int dend = te + ue;
    const int u    = tid;

    int cur = 1;
    for (int d = 1; d <= T_ - 1 + U_; ++d, cur ^= 1) {
        const int t = d - u;
        if (u <= U_ && t >= 0 && t < T_) {
            float va = NEG_INF, vb = NEG_INF;
            if (t > 0) va = alpha[cur ^ 1][u]     + lpb_s[(t - 1) * (U_ + 1) + u];
            if (u > 0) vb = alpha[cur ^ 1][u - 1] + lpl_s[t * U_ + (u - 1)];
            const float mx = fmaxf(va, vb);
            const float mn = fminf(va, vb);
            const float r  = (mx == NEG_INF) ? NEG_INF
                                             : mx + log1pf(__expf(mn - mx));
            alpha[cur][u] = r;
            if (d == dend && u == ue)
                ll[b] = r + lpb_s[te * (U_ + 1) + ue];  // final blank transition
        }
        __syncthreads();
    }
}

// ---------------------------------------------------------------------------
// Phase 1: streaming log-softmax gather.
// One wave32 per (b,t,u) row of 512 floats; 4x b128 nontemporal loads per
// lane (read-once stream must not flush L2, which we want to keep holding the
// 1 MB lpb/lpl workspace for phase 2). Butterfly shuffle reductions for
// max and sum(exp). Only two outputs per row.
// 272 MB / 23.3 TB/s ~= 11.7 us floor; this kernel is pure bandwidth.
// ---------------------------------------------------------------------------
__global__ __launch_bounds__(256)
void rnnt_lse_kernel(const float* __restrict__ logits,
                     const int*   __restrict__ targets,
                     float* __restrict__ lpb,   // [B,T,U+1]
                     float* __restrict__ lpl) { // [B,T,U]
    const int lane = threadIdx.x & 31;
    const int wave = threadIdx.x >> 5;
    const int r    = blockIdx.x * 8 + wave;   // row id in [0, B*T*(U+1))
    const int u    = r % (U_ + 1);
    const int bt   = r / (U_ + 1);            // b*T + t
    const int b    = bt / T_;

    const rnnt_v4f* row4 =
        reinterpret_cast<const rnnt_v4f*>(logits) + (size_t)r * (V_ / 4);
    const rnnt_v4f v0 = __builtin_nontemporal_load(row4 + lane);
    const rnnt_v4f v1 = __builtin_nontemporal_load(row4 + lane + 32);
    const rnnt_v4f v2 = __builtin_nontemporal_load(row4 + lane + 64);
    const rnnt_v4f v3 = __builtin_nontemporal_load(row4 + lane + 96);

    float mx = fmaxf(fmaxf(fmaxf(v0.x, v0.y), fmaxf(v0.z, v0.w)),
                     fmaxf(fmaxf(v1.x, v1.y), fmaxf(v1.z, v1.w)));
    mx = fmaxf(mx, fmaxf(fmaxf(fmaxf(v2.x, v2.y), fmaxf(v2.z, v2.w)),
                         fmaxf(fmaxf(v3.x, v3.y), fmaxf(v3.z, v3.w))));
#pragma unroll
    for (int o = 16; o > 0; o >>= 1) mx = fmaxf(mx, __shfl_xor(mx, o, 32));

    float s = __expf(v0.x - mx) + __expf(v0.y - mx) + __expf(v0.z - mx) + __expf(v0.w - mx)
            + __expf(v1.x - mx) + __expf(v1.y - mx) + __expf(v1.z - mx) + __expf(v1.w - mx)
            + __expf(v2.x - mx) + __expf(v2.y - mx) + __expf(v2.z - mx) + __expf(v2.w - mx)
            + __expf(v3.x - mx) + __expf(v3.y - mx) + __expf(v3.z - mx) + __expf(v3.w - mx);
#pragma unroll
    for (int o = 16; o > 0; o >>= 1) s += __shfl_xor(s, o, 32);

    const float lse = mx + __logf(s);

    if (lane == 0) lpb[r] = v0.x - lse;  // element 0 lives in lane 0 / v0.x

    if (u < U_) {
        const int tgt   = targets[b * U_ + u];  // wave-uniform (scalar load)
        const int chunk = tgt >> 7;             // which of v0..v3
        const int src   = (tgt >> 2) & 31;      // owning lane
        const int comp  = tgt & 3;              // component within float4
        // SCALAR component selects only (uniform conditions -> cndmask/cselect);
        // selecting whole float4 objects made clang build a scratch alloca.
        const bool c1 = (chunk & 1) != 0;
        const bool c2 = (chunk & 2) != 0;
        const float sx = c2 ? (c1 ? v3.x : v2.x) : (c1 ? v1.x : v0.x);
        const float sy = c2 ? (c1 ? v3.y : v2.y) : (c1 ? v1.y : v0.y);
        const float sz = c2 ? (c1 ? v3.z : v2.z) : (c1 ? v1.z : v0.z);
        const float sw = c2 ? (c1 ? v3.w : v2.w) : (c1 ? v1.w : v0.w);
        const bool k1 = (comp & 1) != 0;
        const bool k2 = (comp & 2) != 0;
        float val = k2 ? (k1 ? sw : sz) : (k1 ? sy : sx);
        val = __shfl(val, src, 32);
        if (lane == 0) lpl[bt * U_ + u] = val - lse;
    }
}

// ---------------------------------------------------------------------------
// Phase 3: out = mean(-ll) over B=8.
// ---------------------------------------------------------------------------
__global__ void rnnt_final_kernel(const float* __restrict__ ll,
                                  float* __restrict__ out) {
    if (threadIdx.x == 0 && blockIdx.x == 0) {
        float s = 0.0f;
#pragma unroll
        for (int i = 0; i < B_; ++i) s += ll[i];
        out[0] = -s * (1.0f / (float)B_);
    }
}

extern "C" void kernel_launch(void* const* d_in, const int* in_sizes, int n_in,
                              void* d_out, int out_size, void* d_ws, size_t ws_size,
                              hipStream_t stream) {
    const float* logits  = (const float*)d_in[0];
    const int*   targets = (const int*)d_in[1];
    const int*   loglen  = (const int*)d_in[2];
    const int*   tgtlen  = (const int*)d_in[3];

    float* lpb = (float*)d_ws;                 // B*T*(U+1) floats
    float* lpl = lpb + B_ * T_ * (U_ + 1);     // B*T*U floats
    float* ll  = lpl + B_ * T_ * U_;           // B floats
    float* out = (float*)d_out;

    const int rows = B_ * T_ * (U_ + 1);       // 133,120 rows; 8 rows per block
    rnnt_lse_kernel<<<rows / 8, 256, 0, stream>>>(logits, targets, lpb, lpl);
    rnnt_dp_kernel<<<B_, 128, 0, stream>>>(lpb, lpl, loglen, tgtlen, ll);
    rnnt_final_kernel<<<1, 32, 0, stream>>>(ll, out);
}